// MarkovKalmanModule_26620207301249
// MI455X (gfx1250) — compile-verified
//
#include <hip/hip_runtime.h>

// CDNA5 / gfx1250 batched Kalman filter step.
// One 512-thread block (16 wave32) per batch element; FP32 WMMA 16x16x4 for
// all 64x64x64 matmuls; pivot-free Gauss-Jordan inversion of the SPD
// innovation covariance in LDS.

typedef __attribute__((ext_vector_type(2))) float v2f;
typedef __attribute__((ext_vector_type(8))) float v8f;

static constexpr int kS  = 64;   // state dim
static constexpr int kLD = 68;   // LDS row stride (floats): 68%64=4 -> conflict-free columns
static constexpr int kNT = 512;  // threads per block (16 waves)

// One 16x16 tile of a 64x64x64 matmul: OUT(tm,tn) += X * Y.
// Element X(i,k) = X[i*xr + k*xc], Y(k,n) = Y[k*yr + n*yc]  (stride swap = transpose).
__device__ __forceinline__ v8f mm_tile(const float* __restrict__ X, int xr, int xc,
                                       const float* __restrict__ Y, int yr, int yc,
                                       int tm, int tn, int lane)
{
  const int m  = tm * 16 + (lane & 15);   // A-frag row (lanes 0-15 / 16-31 both M=0..15)
  const int nn = tn * 16 + (lane & 15);   // B-frag col
  const int kh = (lane >> 4) * 2;         // lanes 16-31 hold K=+2,+3
  v8f acc = {0.f, 0.f, 0.f, 0.f, 0.f, 0.f, 0.f, 0.f};
#pragma unroll
  for (int k0 = 0; k0 < kS; k0 += 4) {
    const int ka = k0 + kh;
    v2f a, b;
    a.x = X[m * xr + ka * xc];
    a.y = X[m * xr + (ka + 1) * xc];
    b.x = Y[ka * yr + nn * yc];
    b.y = Y[(ka + 1) * yr + nn * yc];
    // 8 args: (neg_a, A, neg_b, B, c_mod, C, reuse_a, reuse_b)
    acc = __builtin_amdgcn_wmma_f32_16x16x4_f32(false, a, false, b, (short)0, acc,
                                                false, false);
  }
  return acc;
}

// C/D layout: VGPR r, lanes 0-15 -> row tm*16+r, lanes 16-31 -> row tm*16+r+8; col = tn*16+(lane&15)
__device__ __forceinline__ void store_tile(float* __restrict__ s, v8f acc,
                                           int tm, int tn, int lane)
{
  const int col = tn * 16 + (lane & 15);
  const int rb  = tm * 16 + (lane >> 4) * 8;
#pragma unroll
  for (int r = 0; r < 8; ++r) s[(rb + r) * kLD + col] = acc[r];
}

__device__ __forceinline__ void store_tile_diag(float* __restrict__ s, v8f acc,
                                                const float* __restrict__ diag,
                                                int tm, int tn, int lane)
{
  const int col = tn * 16 + (lane & 15);
  const int rb  = tm * 16 + (lane >> 4) * 8;
#pragma unroll
  for (int r = 0; r < 8; ++r) {
    const int row = rb + r;
    float v = acc[r];
    if (row == col) v += diag[row];
    s[row * kLD + col] = v;
  }
}

// Copy a row-major 64x64 global matrix into LDS with row stride kLD (float4 path).
__device__ __forceinline__ void load_mat(const float* __restrict__ g,
                                         float* __restrict__ s, int tid)
{
  for (int idx = tid; idx < kS * kS / 4; idx += kNT) {  // 1024 float4s
    const int row = idx >> 4;                           // 16 float4 per row
    const int c4  = idx & 15;
    const float4 v = ((const float4*)g)[idx];
    *(float4*)(s + row * kLD + c4 * 4) = v;
  }
}

__global__ __launch_bounds__(kNT)
void kalman_step_kernel(const float* __restrict__ x_post,
                        const float* __restrict__ P_post,
                        const float* __restrict__ meas,
                        const float* __restrict__ A,
                        const float* __restrict__ H,
                        const float* __restrict__ logQ,
                        const float* __restrict__ logR,
                        float* __restrict__ out_x,
                        float* __restrict__ out_P,
                        float* __restrict__ out_innov)
{
  extern __shared__ float smem[];
  float* sA  = smem;               // A
  float* sH  = sA + kS * kLD;      // H
  float* sB0 = sH + kS * kLD;      // P_post -> Smat -> Sinv
  float* sB1 = sB0 + kS * kLD;     // T1 = A*P -> K
  float* sB2 = sB1 + kS * kLD;     // P_pred
  float* sB3 = sB2 + kS * kLD;     // T2 = H*P_pred
  float* sv  = sB3 + kS * kLD;     // [0:64) x_post | [64:128) x_pred | [128:192) z
                                   // [192:256) innov | [256:320) expQ | [320:384) expR+eps

  const int tid  = threadIdx.x;
  const int lane = tid & 31;
  const int wave = tid >> 5;
  const int tm   = wave >> 2;
  const int tn   = wave & 3;
  const int b    = blockIdx.x;

  // ---- stage inputs into LDS ----
  load_mat(A, sA, tid);
  load_mat(H, sH, tid);
  load_mat(P_post + (size_t)b * kS * kS, sB0, tid);
  if (tid < kS) {
    sv[tid]       = x_post[(size_t)b * kS + tid];
    sv[128 + tid] = meas[(size_t)b * kS + tid];
    sv[256 + tid] = __expf(logQ[tid]);
    sv[320 + tid] = __expf(logR[tid]) + 1e-6f;
  }
  __syncthreads();

  // ---- x_pred = A x_post ----
  if (tid < kS) {
    float acc = 0.f;
    for (int j = 0; j < kS; ++j) acc += sA[tid * kLD + j] * sv[j];
    sv[64 + tid] = acc;
  }
  __syncthreads();

  // ---- innovation = z - H x_pred (waves 0-1) overlapped with T1 = A * P_post ----
  if (tid < kS) {
    float acc = 0.f;
    for (int j = 0; j < kS; ++j) acc += sH[tid * kLD + j] * sv[64 + j];
    sv[192 + tid] = sv[128 + tid] - acc;
  }
  store_tile(sB1, mm_tile(sA, kLD, 1, sB0, kLD, 1, tm, tn, lane), tm, tn, lane);
  __syncthreads();

  // ---- P_pred = T1 * A^T + diag(expQ) ----
  store_tile_diag(sB2, mm_tile(sB1, kLD, 1, sA, 1, kLD, tm, tn, lane),
                  sv + 256, tm, tn, lane);
  __syncthreads();

  // ---- T2 = H * P_pred ----
  store_tile(sB3, mm_tile(sH, kLD, 1, sB2, kLD, 1, tm, tn, lane), tm, tn, lane);
  __syncthreads();

  // ---- Smat = T2 * H^T + diag(expR + 1e-6) ----
  store_tile_diag(sB0, mm_tile(sB3, kLD, 1, sH, 1, kLD, tm, tn, lane),
                  sv + 320, tm, tn, lane);
  __syncthreads();

  // ---- In-place Gauss-Jordan inversion of SPD Smat (no pivoting needed) ----
  {
    const int gi = tid >> 3;          // row handled by this thread (8 cols each)
    const int j0 = (tid & 7) * 8;
    for (int k = 0; k < kS; ++k) {
      const float dinv = 1.0f / sB0[k * kLD + k];
      if (tid < kS && tid != k) sB0[k * kLD + tid] *= dinv;  // scale pivot row
      __syncthreads();
      if (gi != k) {
        const float f = sB0[gi * kLD + k];
#pragma unroll
        for (int t = 0; t < 8; ++t) {
          const int j = j0 + t;
          if (j != k) sB0[gi * kLD + j] -= f * sB0[k * kLD + j];
        }
      }
      __syncthreads();
      if (tid < kS) {
        if (tid != k) sB0[tid * kLD + k] = -sB0[tid * kLD + k] * dinv;
        else          sB0[k * kLD + k]  = dinv;
      }
      __syncthreads();
    }
  }

  // ---- K = (P_pred H^T) Sinv = T2^T * Sinv  (P_pred symmetric) ----
  store_tile(sB1, mm_tile(sB3, 1, kLD, sB0, kLD, 1, tm, tn, lane), tm, tn, lane);
  __syncthreads();

  // ---- P_new = P_pred - K * T2, streamed straight to global ----
  {
    const v8f kt2 = mm_tile(sB1, kLD, 1, sB3, kLD, 1, tm, tn, lane);
    const int col = tn * 16 + (lane & 15);
    const int rb  = tm * 16 + (lane >> 4) * 8;
    float* outPb = out_P + (size_t)b * kS * kS;
#pragma unroll
    for (int r = 0; r < 8; ++r) {
      const int row = rb + r;
      outPb[row * kS + col] = sB2[row * kLD + col] - kt2[r];
    }
  }

  // ---- x_new = x_pred + K innov ; emit innovation ----
  if (tid < kS) {
    float acc = 0.f;
    for (int j = 0; j < kS; ++j) acc += sB1[tid * kLD + j] * sv[192 + j];
    out_x[(size_t)b * kS + tid]     = sv[64 + tid] + acc;
    out_innov[(size_t)b * kS + tid] = sv[192 + tid];
  }
}

extern "C" void kernel_launch(void* const* d_in, const int* in_sizes, int n_in,
                              void* d_out, int out_size, void* d_ws, size_t ws_size,
                              hipStream_t stream)
{
  const float* x_post = (const float*)d_in[0];
  const float* P_post = (const float*)d_in[1];
  const float* meas   = (const float*)d_in[2];
  const float* A      = (const float*)d_in[3];
  const float* H      = (const float*)d_in[4];
  const float* logQ   = (const float*)d_in[5];
  const float* logR   = (const float*)d_in[6];

  const int B = in_sizes[0] / kS;  // 16384

  float* out       = (float*)d_out;
  float* out_x     = out;                              // (B,S)
  float* out_P     = out_x + (size_t)B * kS;           // (B,S,S)
  float* out_innov = out_P + (size_t)B * kS * kS;      // (B,S)

  const size_t lds_bytes = (size_t)(6 * kS * kLD + 384) * sizeof(float);

  kalman_step_kernel<<<dim3(B), dim3(kNT), lds_bytes, stream>>>(
      x_post, P_post, meas, A, H, logQ, logR, out_x, out_P, out_innov);
}